// GIN_8426725834832
// MI455X (gfx1250) — compile-verified
//
#include <hip/hip_runtime.h>
#include <hip/hip_bf16.h>

typedef float v2f __attribute__((ext_vector_type(2)));
typedef float v8f __attribute__((ext_vector_type(8)));

#define N_NODES 100000
#define N_EDGES 1000000
#define N_GRAPHS 2000

// hardware f32 atomic add, no-return form (STOREcnt), device scope
__device__ __forceinline__ void atomAddF32(float* p, float v) {
    asm volatile("global_atomic_add_f32 %0, %1, off scope:SCOPE_DEV"
                 :: "v"((unsigned long long)p), "v"(v)
                 : "memory");
}

__device__ __forceinline__ float leaky1(float v) { return v > 0.0f ? v : 0.01f * v; }

// ---------------------------------------------------------------------------
// zero fill
__global__ void zero_f32(float* __restrict__ p, long n) {
    long i = (long)blockIdx.x * blockDim.x + threadIdx.x;
    long s = (long)gridDim.x * blockDim.x;
    for (; i < n; i += s) p[i] = 0.0f;
}

// ---------------------------------------------------------------------------
// scatter-add: agg[dst] += x[src], thread = (edge, 4-dim chunk)
__global__ void scatter_add(const float* __restrict__ xin, int xstride,
                            const int* __restrict__ src, const int* __restrict__ dst,
                            float* __restrict__ agg, int nEdges) {
    long t = (long)blockIdx.x * blockDim.x + threadIdx.x;
    if (t >= (long)nEdges * 16) return;
    int e = (int)(t >> 4);
    int d = (int)(t & 15) * 4;
    int s  = src[e];
    int de = dst[e];
    const float4 v = *(const float4*)(xin + (long)s * xstride + d);
    float* a = agg + (long)de * 64 + d;
    atomAddF32(a + 0, v.x);
    atomAddF32(a + 1, v.y);
    atomAddF32(a + 2, v.z);
    atomAddF32(a + 3, v.w);
}

// ---------------------------------------------------------------------------
// GIN layer: out[:, l*64:(l+1)*64] = leaky(leaky(BN(((1+eps)x + agg) @ W + b)))
// block = 256 threads = 8 waves; each wave computes a 16-node x 64-col tile
// with v_wmma_f32_16x16x4_f32 (fp32-exact).
__global__ __launch_bounds__(256) void gin_layer(
    const float* __restrict__ xin, int xstride,
    const float* __restrict__ agg,
    const float* __restrict__ W,      // [64,64] row-major (k-major)
    const float* __restrict__ bias,   // [64]
    const float* __restrict__ gamma, const float* __restrict__ beta,
    const float* __restrict__ mean,  const float* __restrict__ var,
    const float* __restrict__ epsArr, int layer,
    float* __restrict__ out,          // node stride 256, cols 0..63 of slice
    int nNodes)
{
    __shared__ float Wt[64 * 64];          // transposed: Wt[n*64 + k]
    __shared__ float atile[8][16 * 64];    // per-wave input tile
    __shared__ float scaleS[64], shiftS[64];

    const int tid = threadIdx.x;
    const float epsl = 1.0f + epsArr[layer];

    // stage W transposed (coalesced global reads)
#pragma unroll
    for (int j = 0; j < 16; ++j) {
        int idx = j * 256 + tid;
        int k = idx >> 6, n = idx & 63;
        Wt[n * 64 + k] = W[idx];
    }
    if (tid < 64) {
        float s = gamma[tid] * rsqrtf(var[tid] + 1e-5f);
        scaleS[tid] = s;
        shiftS[tid] = (bias[tid] - mean[tid]) * s + beta[tid];
    }

    const int wv   = tid >> 5;
    const int lane = tid & 31;
    const int half = lane >> 4;
    const int l16  = lane & 15;
    const long row0 = (long)blockIdx.x * 128 + wv * 16;

    // stage combined input (1+eps)*x + agg -> LDS (zeros past end)
#pragma unroll
    for (int j = 0; j < 8; ++j) {
        int idx = j * 128 + lane * 4;
        int r = idx >> 6, c = idx & 63;
        long grow = row0 + r;
        float4 v = make_float4(0.f, 0.f, 0.f, 0.f);
        if (grow < nNodes) {
            float4 xv = *(const float4*)(xin + grow * (long)xstride + c);
            float4 av = *(const float4*)(agg + grow * 64 + c);
            v.x = epsl * xv.x + av.x;
            v.y = epsl * xv.y + av.y;
            v.z = epsl * xv.z + av.z;
            v.w = epsl * xv.w + av.w;
        }
        *(float4*)&atile[wv][idx] = v;
    }
    __syncthreads();

    v8f z = {0.f, 0.f, 0.f, 0.f, 0.f, 0.f, 0.f, 0.f};
    v8f acc[4] = {z, z, z, z};
    const float* at = atile[wv];
#pragma unroll
    for (int k = 0; k < 64; k += 4) {
        int kk = k + half * 2;
        v2f a = *(const v2f*)&at[l16 * 64 + kk];
#pragma unroll
        for (int nt = 0; nt < 4; ++nt) {
            v2f b = *(const v2f*)&Wt[(nt * 16 + l16) * 64 + kk];
            acc[nt] = __builtin_amdgcn_wmma_f32_16x16x4_f32(
                false, a, false, b, (short)0, acc[nt], false, false);
        }
    }

    // epilogue: folded bias+BN, double leaky, write into 256-stride concat slice
#pragma unroll
    for (int nt = 0; nt < 4; ++nt) {
        int col = nt * 16 + l16;
        float s  = scaleS[col];
        float sh = shiftS[col];
#pragma unroll
        for (int i = 0; i < 8; ++i) {
            long grow = row0 + i + half * 8;
            if (grow < nNodes) {
                float v = acc[nt][i] * s + sh;
                v = leaky1(v);
                v = leaky1(v);
                out[grow * 256 + col] = v;
            }
        }
    }
}

// ---------------------------------------------------------------------------
// global add pool: pool[batch[n]] += h3[n]
__global__ void pool_accum(const float* __restrict__ h3, const int* __restrict__ batch,
                           float* __restrict__ pool, int nNodes) {
    long t = (long)blockIdx.x * blockDim.x + threadIdx.x;
    if (t >= (long)nNodes * 16) return;
    int n = (int)(t >> 4);
    int d = (int)(t & 15) * 4;
    int g = batch[n];
    float4 v = *(const float4*)(h3 + (long)n * 256 + d);
    float* p = pool + (long)g * 64 + d;
    atomAddF32(p + 0, v.x);
    atomAddF32(p + 1, v.y);
    atomAddF32(p + 2, v.z);
    atomAddF32(p + 3, v.w);
}

// broadcast pool back per node into concat slice [192:256]
__global__ void pool_gather(const float* __restrict__ pool, const int* __restrict__ batch,
                            float* __restrict__ hcat, int nNodes) {
    long t = (long)blockIdx.x * blockDim.x + threadIdx.x;
    if (t >= (long)nNodes * 16) return;
    int n = (int)(t >> 4);
    int d = (int)(t & 15) * 4;
    int g = batch[n];
    float4 v = *(const float4*)(pool + (long)g * 64 + d);
    *(float4*)(hcat + (long)n * 256 + 192 + d) = v;
}

// ---------------------------------------------------------------------------
// fused classifier: [N,256] -> leaky(@W0+b0) -> leaky(@W1+b1) -> @W2+b2 -> sigmoid
// one wave per block, TWO 16-node tiles per wave (B fragments reused across both
// -> halves W0/W1 L2 traffic). A tiles staged with global_load_async_to_lds_b128.
// LDS: atile 32KB + h0 32KB = 64KB; h1 aliases the dead atile region.
__global__ __launch_bounds__(32) void classifier(
    const float* __restrict__ hcat,
    const float* __restrict__ W0, const float* __restrict__ b0,   // [256,256],[256]
    const float* __restrict__ W1, const float* __restrict__ b1,   // [256,128],[128]
    const float* __restrict__ W2, const float* __restrict__ b2,   // [128],[1]
    float* __restrict__ outp, int nNodes)
{
    __shared__ float smem[16384];          // 64 KB
    float* const atile = smem;             // [32 x 256] (dead after GEMM0)
    float* const h0    = smem + 8192;      // [32 x 256]
    float* const h1t   = smem;             // [32 x 128] aliases atile

    const int lane = threadIdx.x;
    const int half = lane >> 4;
    const int l16  = lane & 15;
    const long row0 = (long)blockIdx.x * 32;

    // stage 32 rows x 256 cols of concat features, async global->LDS
#pragma unroll 4
    for (int j = 0; j < 64; ++j) {
        int c4 = j * 32 + lane;            // float4 index within 32x256 block
        int r  = c4 >> 6;                  // 0..31
        int c  = (c4 & 63) * 4;
        long grow = row0 + r;
        unsigned ldsaddr = (unsigned)(unsigned long long)&atile[r * 256 + c];
        if (grow < nNodes) {
            unsigned long long ga = (unsigned long long)(hcat + grow * 256 + c);
            asm volatile("global_load_async_to_lds_b128 %0, %1, off"
                         :: "v"(ldsaddr), "v"(ga) : "memory");
        } else {
            *(float4*)&atile[r * 256 + c] = make_float4(0.f, 0.f, 0.f, 0.f);
        }
    }
    asm volatile("s_wait_asynccnt 0x0" ::: "memory");
    __syncthreads();

    const v8f z = {0.f, 0.f, 0.f, 0.f, 0.f, 0.f, 0.f, 0.f};

    // GEMM0: 256 -> 256, two node tiles share each B fragment
    for (int nt = 0; nt < 16; ++nt) {
        v8f acc0 = z, acc1 = z;
        int col = nt * 16 + l16;
        for (int k = 0; k < 256; k += 4) {
            int kk = k + half * 2;
            v2f b;
            b.x = W0[(long)kk * 256 + col];
            b.y = W0[(long)(kk + 1) * 256 + col];
            v2f a0 = *(const v2f*)&atile[l16 * 256 + kk];
            v2f a1 = *(const v2f*)&atile[(16 + l16) * 256 + kk];
            acc0 = __builtin_amdgcn_wmma_f32_16x16x4_f32(
                false, a0, false, b, (short)0, acc0, false, false);
            acc1 = __builtin_amdgcn_wmma_f32_16x16x4_f32(
                false, a1, false, b, (short)0, acc1, false, false);
        }
        float bb = b0[col];
#pragma unroll
        for (int i = 0; i < 8; ++i) {
            int r = i + half * 8;
            h0[r * 256 + col]        = leaky1(acc0[i] + bb);
            h0[(16 + r) * 256 + col] = leaky1(acc1[i] + bb);
        }
    }
    __syncthreads();   // atile dead from here; h1t may overwrite it

    // GEMM1: 256 -> 128
    for (int nt = 0; nt < 8; ++nt) {
        v8f acc0 = z, acc1 = z;
        int col = nt * 16 + l16;
        for (int k = 0; k < 256; k += 4) {
            int kk = k + half * 2;
            v2f b;
            b.x = W1[(long)kk * 128 + col];
            b.y = W1[(long)(kk + 1) * 128 + col];
            v2f a0 = *(const v2f*)&h0[l16 * 256 + kk];
            v2f a1 = *(const v2f*)&h0[(16 + l16) * 256 + kk];
            acc0 = __builtin_amdgcn_wmma_f32_16x16x4_f32(
                false, a0, false, b, (short)0, acc0, false, false);
            acc1 = __builtin_amdgcn_wmma_f32_16x16x4_f32(
                false, a1, false, b, (short)0, acc1, false, false);
        }
        float bb = b1[col];
#pragma unroll
        for (int i = 0; i < 8; ++i) {
            int r = i + half * 8;
            h1t[r * 128 + col]        = leaky1(acc0[i] + bb);
            h1t[(16 + r) * 128 + col] = leaky1(acc1[i] + bb);
        }
    }
    __syncthreads();

    // final 128 -> 1 dot + sigmoid: lanes 0..15 -> tile0 rows, 16..31 -> tile1
    {
        int r = half * 16 + l16;
        long grow = row0 + r;
        if (grow < nNodes) {
            float s = b2[0];
            const float* hrow = &h1t[r * 128];
#pragma unroll 8
            for (int j = 0; j < 128; ++j) s += hrow[j] * W2[j];
            outp[grow] = 1.0f / (1.0f + expf(-s));
        }
    }
}

// ---------------------------------------------------------------------------
extern "C" void kernel_launch(void* const* d_in, const int* in_sizes, int n_in,
                              void* d_out, int out_size, void* d_ws, size_t ws_size,
                              hipStream_t stream) {
    const float* x     = (const float*)d_in[0];
    const int*   edge  = (const int*)d_in[1];   // [2, E]
    const int*   batch = (const int*)d_in[2];
    const float* convW = (const float*)d_in[3]; // [3,64,64]
    const float* convb = (const float*)d_in[4]; // [3,64]
    const float* gamma = (const float*)d_in[5];
    const float* beta  = (const float*)d_in[6];
    const float* mean  = (const float*)d_in[7];
    const float* var   = (const float*)d_in[8];
    const float* eps   = (const float*)d_in[9]; // [3]
    const float* W0    = (const float*)d_in[10];
    const float* b0    = (const float*)d_in[11];
    const float* W1    = (const float*)d_in[12];
    const float* b1    = (const float*)d_in[13];
    const float* W2    = (const float*)d_in[14];
    const float* b2    = (const float*)d_in[15];
    float* out = (float*)d_out;

    const int N = N_NODES, E = N_EDGES, G = N_GRAPHS;
    float* ws   = (float*)d_ws;
    float* agg  = ws;                         // N*64
    float* hcat = ws + (size_t)N * 64;        // N*256 concat buffer
    float* pool = hcat + (size_t)N * 256;     // G*64
    const int* src = edge;
    const int* dst = edge + E;

    const int scatterBlocks = (int)(((long)E * 16 + 255) / 256);
    const int nodeVecBlocks = (int)(((long)N * 16 + 255) / 256);
    const int ginBlocks     = (N + 127) / 128;

    for (int l = 0; l < 3; ++l) {
        zero_f32<<<2048, 256, 0, stream>>>(agg, (long)N * 64);
        const float* xin = (l == 0) ? x : (hcat + (l - 1) * 64);
        int xs = (l == 0) ? 64 : 256;
        scatter_add<<<scatterBlocks, 256, 0, stream>>>(xin, xs, src, dst, agg, E);
        gin_layer<<<ginBlocks, 256, 0, stream>>>(
            xin, xs, agg,
            convW + (size_t)l * 64 * 64, convb + (size_t)l * 64,
            gamma + (size_t)l * 64, beta + (size_t)l * 64,
            mean + (size_t)l * 64, var + (size_t)l * 64,
            eps, l, hcat + (size_t)l * 64, N);
    }

    zero_f32<<<256, 256, 0, stream>>>(pool, (long)G * 64);
    pool_accum<<<nodeVecBlocks, 256, 0, stream>>>(hcat + 128, batch, pool, N);
    pool_gather<<<nodeVecBlocks, 256, 0, stream>>>(pool, batch, hcat, N);

    classifier<<<(N + 31) / 32, 32, 0, stream>>>(hcat, W0, b0, W1, b1, W2, b2, out, N);
}